// Model_69965017252507
// MI455X (gfx1250) — compile-verified
//
#include <hip/hip_runtime.h>
#include <hip/hip_bf16.h>

// ---------------------------------------------------------------------------
// DLinear fused forecast for MI455X (gfx1250, wave32, WMMA).
//
// Reference math:   trend = MA_k(x) (edge pad), seasonal = x - trend
//                   out = W_t@trend + W_s@seasonal + b_t + b_s
// Fused form:       out[b] = W_eff @ x[b] + bias,
//   W_eff[p,s] = W_s[p,s] + (1/24)*( sum_{t=s-12..s+11, clamped} Wd[p,t]
//                                    + [s==0]   * sum_{t<11}   (11-t)   Wd[p,t]
//                                    + [s==1023]* sum_{t>=1012}(t-1011) Wd[p,t] )
//   Wd = W_t - W_s,  bias = b_t + b_s
// ---------------------------------------------------------------------------

typedef __attribute__((ext_vector_type(2))) float v2f;
typedef __attribute__((ext_vector_type(8))) float v8f;

#define B_      128
#define S_      1024
#define C_      64
#define P_      336
#define K_MA    24
#define PADL    11   // (k-1)/2
#define PADR    12   // k/2
#define SCHUNK  128  // s-chunk staged in LDS: 128*64*4B = 32 KB
#define WAVES   7    // waves per workgroup -> 7 P-tiles of 16 rows

// ---------------- prep: build W_eff [P,S] and bias [P] -----------------------
__global__ __launch_bounds__(256)
void weff_prep_kernel(const float* __restrict__ wt,
                      const float* __restrict__ ws,
                      const float* __restrict__ bt,
                      const float* __restrict__ bs,
                      float* __restrict__ weff,
                      float* __restrict__ bias)
{
    int idx = blockIdx.x * blockDim.x + threadIdx.x;
    if (idx >= P_ * S_) return;
    int p = idx / S_;
    int s = idx - p * S_;
    const float* wtr = wt + (size_t)p * S_;
    const float* wsr = ws + (size_t)p * S_;

    int lo = s - PADR; if (lo < 0) lo = 0;
    int hi = s + PADL; if (hi > S_ - 1) hi = S_ - 1;
    float sum = 0.f;
    for (int t = lo; t <= hi; ++t) sum += wtr[t] - wsr[t];

    if (s == 0) {
        // clamped-left multiplicity: x[0] appears (11 - t) extra times in window t
        for (int t = 0; t < PADL; ++t)
            sum += (float)(PADL - t) * (wtr[t] - wsr[t]);
        bias[p] = bt[p] + bs[p];
    }
    if (s == S_ - 1) {
        // clamped-right multiplicity: x[S-1] appears (t - 1011) extra times
        for (int t = S_ - PADR; t < S_; ++t)
            sum += (float)(t - (S_ - 1 - PADR)) * (wtr[t] - wsr[t]);
    }
    weff[idx] = wsr[s] + sum * (1.0f / (float)K_MA);
}

// ---------------- fused GEMM: out[b] = W_eff @ x[b] + bias -------------------
// grid = (3, 128); block = 224 (7 waves). Wave w of group g owns P-tile
// (g*7 + w): rows p0..p0+15, all 64 channels, K = 1024 in steps of 4 via
// v_wmma_f32_16x16x4_f32 (full fp32 precision; problem is HBM-bound anyway).
//
// WMMA f32 16x16x4 register layouts (ISA 7.12.2):
//   A 16x4 : lane L holds row M=L%16; VGPR v holds K = 2*(L/16) + v
//   B 4x16 : lane L holds col N=L%16; VGPR v holds K = 2*(L/16) + v (mirror)
//   C/D    : VGPR v: lanes 0-15 -> M=v, lanes 16-31 -> M=v+8; N = L%16
__global__ __launch_bounds__(WAVES * 32)
void dlinear_wmma_kernel(const float* __restrict__ x,
                         const float* __restrict__ weff,
                         const float* __restrict__ bias,
                         float* __restrict__ out)
{
    __shared__ float lds_x[SCHUNK * C_];  // 32 KB staging of x[b] chunk

    const int tid   = threadIdx.x;
    const int lane  = tid & 31;
    const int wave  = tid >> 5;
    const int b     = blockIdx.y;
    const int p0    = (blockIdx.x * WAVES + wave) * 16;
    const int half  = lane >> 4;   // 0: lanes 0-15, 1: lanes 16-31
    const int lid   = lane & 15;

    // Seed accumulators with bias[p] (broadcast across the 16 N columns).
    v8f acc[4];
#pragma unroll
    for (int v = 0; v < 8; ++v) {
        float bv = bias[p0 + v + 8 * half];
        acc[0][v] = bv; acc[1][v] = bv; acc[2][v] = bv; acc[3][v] = bv;
    }

    const float* xb   = x + (size_t)b * S_ * C_;
    // A-row base for this lane: row (p0+lid), K offset 2*half (b64 per k-step)
    const float* wrow = weff + (size_t)(p0 + lid) * S_ + 2 * half;

    for (int chunk = 0; chunk < S_ / SCHUNK; ++chunk) {
        // Cooperative 128B-wide staging of x chunk into LDS.
        const float4* src = (const float4*)(xb + chunk * SCHUNK * C_);
        float4*       dst = (float4*)lds_x;
        for (int i = tid; i < SCHUNK * C_ / 4; i += WAVES * 32)
            dst[i] = src[i];
        __syncthreads();

        // Pull the next chunk toward L2 while we compute on this one.
        if (chunk + 1 < S_ / SCHUNK) {
            const char* nxt = (const char*)(xb + (chunk + 1) * SCHUNK * C_);
            __builtin_prefetch(nxt + (size_t)tid * 128, 0, 1);
        }

        const float* wk = wrow + chunk * SCHUNK;
#pragma unroll 4
        for (int ks = 0; ks < SCHUNK / 4; ++ks) {
            // A tile (16x4 of W_eff): two K-consecutive floats per lane.
            v2f a = *(const v2f*)(wk + ks * 4);
            const float* bb = lds_x + (ks * 4 + 2 * half) * C_;
#pragma unroll
            for (int t = 0; t < 4; ++t) {
                v2f bm;
                bm.x = bb[t * 16 + lid];        // K = 2*half,   N = t*16+lid
                bm.y = bb[C_ + t * 16 + lid];   // K = 2*half+1
                acc[t] = __builtin_amdgcn_wmma_f32_16x16x4_f32(
                    false, a, false, bm, (short)0, acc[t], false, false);
            }
        }
        __syncthreads();
    }

    // Epilogue: scatter D tiles. Lanes 0-15 / 16-31 cover rows v / v+8.
    float* ob = out + (size_t)b * P_ * C_;
#pragma unroll
    for (int t = 0; t < 4; ++t)
#pragma unroll
        for (int v = 0; v < 8; ++v)
            ob[(size_t)(p0 + v + 8 * half) * C_ + t * 16 + lid] = acc[t][v];
}

// ---------------------------------------------------------------------------
extern "C" void kernel_launch(void* const* d_in, const int* in_sizes, int n_in,
                              void* d_out, int out_size, void* d_ws, size_t ws_size,
                              hipStream_t stream) {
    (void)in_sizes; (void)n_in; (void)out_size; (void)ws_size;
    const float* x  = (const float*)d_in[0];
    const float* wt = (const float*)d_in[1];
    const float* bt = (const float*)d_in[2];
    const float* ws = (const float*)d_in[3];
    const float* bs = (const float*)d_in[4];
    float* out  = (float*)d_out;
    float* weff = (float*)d_ws;               // P_*S_ floats (~1.38 MB)
    float* bias = weff + (size_t)P_ * S_;     // P_ floats

    int n = P_ * S_;
    weff_prep_kernel<<<(n + 255) / 256, 256, 0, stream>>>(wt, ws, bt, bs, weff, bias);

    dim3 grid(P_ / (16 * WAVES), B_);         // (3, 128)
    dlinear_wmma_kernel<<<grid, WAVES * 32, 0, stream>>>(x, weff, bias, out);
}